// RandomShiftsAug_41575283425426
// MI455X (gfx1250) — compile-verified
//
#include <hip/hip_runtime.h>
#include <stdint.h>

// RandomShiftsAug on 1x1 spatial maps degenerates to a per-row mask:
//   out[i,:] = x[i,:] iff (h1[i]==pad && w1[i]==pad), else 0.
// Pure store-bandwidth bound (~268 MB writes, ~3 MB reads after scalar
// row-skip -> ~11.6 us at 23.3 TB/s). No matmul -> no WMMA; the CDNA5
// features that fit are the async LDS mover (ASYNCcnt path) for the ~1.2%
// of rows that are copied, and NT b128 stores for the zero stream.

typedef float f4 __attribute__((ext_vector_type(4)));
typedef __attribute__((address_space(3))) float lds_float;

#define THREADS 256
#define TILEV   1024   // 1024 float4 vecs = 16 KB LDS tile (one full row at C=4096)

__global__ __launch_bounds__(THREADS)
void RandomShiftsAug_kernel(const float* __restrict__ x,
                            const int* __restrict__ h1,
                            const int* __restrict__ w1,
                            const int* __restrict__ pad_p,
                            float* __restrict__ out,
                            int cvec /* C/4 */) {
    // b128 async LDS transfers require 16-byte-aligned LDS addresses.
    __shared__ __align__(16) float lds_buf[TILEV * 4];   // 16 KB staging tile

    const int row = blockIdx.x;
    const int pad = pad_p[0];
    // Uniform per-block condition -> scalar SMEM loads + scalar branch
    // (verified in round-1 asm: s_load_b32 + s_cmp + s_cbranch).
    const int keep = __builtin_amdgcn_readfirstlane(
        (h1[row] == pad && w1[row] == pad) ? 1 : 0);

    const size_t base = (size_t)row * (size_t)cvec * 4u;

    if (!keep) {
        // Dominant path (~98.8% of rows): stream zeros, 128-bit NT stores,
        // no x read at all. Unrolled -> 4 back-to-back global_store_b128
        // with immediate offsets at C=4096.
        f4 z = {0.0f, 0.0f, 0.0f, 0.0f};
        f4* o = (f4*)(out + base);
#pragma unroll 4
        for (int v = threadIdx.x; v < cvec; v += THREADS) {
            __builtin_nontemporal_store(z, &o[v]);
        }
    } else {
        // Rare path: bulk-copy the row via the gfx1250 async LDS mover.
        // global -> LDS -> global, no VGPR data movement, ASYNCcnt-tracked.
        const float* gin  = x   + base;   // uniform -> SGPR pair (saddr)
        float*       gout = out + base;   // uniform -> SGPR pair (saddr)
        // addrspace(3) pointer: low 32 bits of the flat shared address are
        // the LDS byte offset (aperture lives in bits 63:32).
        lds_float* lbuf = (lds_float*)(uintptr_t)&lds_buf[0];

        for (int v0 = 0; v0 < cvec; v0 += TILEV) {
            const int nv = (cvec - v0 < TILEV) ? (cvec - v0) : TILEV;

            // Phase 1: async load tile into LDS (each lane moves 16 B/iter;
            // unrolled -> up to 4 loads in flight per wave, ASYNCcnt <= 4).
#pragma unroll 4
            for (int v = threadIdx.x; v < nv; v += THREADS) {
                unsigned goff = (unsigned)(v0 + v) * 16u;  // byte off vs saddr
                asm volatile(
                    "global_load_async_to_lds_b128 %0, %1, %2"
                    :
                    : "v"(lbuf + (size_t)v * 4u), "v"(goff), "s"(gin)
                    : "memory");
            }
            // Wait for this wave's async loads to land in LDS. Each wave
            // only stores the chunks it loaded itself, so no barrier needed.
            asm volatile("s_wait_asynccnt 0" ::: "memory");

            // Phase 2: async store tile from LDS back to global.
#pragma unroll 4
            for (int v = threadIdx.x; v < nv; v += THREADS) {
                unsigned goff = (unsigned)(v0 + v) * 16u;
                asm volatile(
                    "global_store_async_from_lds_b128 %0, %1, %2"
                    :
                    : "v"(goff), "v"(lbuf + (size_t)v * 4u), "s"(gout)
                    : "memory");
            }
            // LDS tile re-use in the next iteration requires the stores to
            // have consumed it.
            asm volatile("s_wait_asynccnt 0" ::: "memory");
        }
        // S_ENDPGM performs an implicit wait-idle, covering any stragglers.
    }
}

extern "C" void kernel_launch(void* const* d_in, const int* in_sizes, int n_in,
                              void* d_out, int out_size, void* d_ws, size_t ws_size,
                              hipStream_t stream) {
    const float* x    = (const float*)d_in[0];
    const int*   h1   = (const int*)d_in[1];
    const int*   w1   = (const int*)d_in[2];
    const int*   padp = (const int*)d_in[3];
    float*       out  = (float*)d_out;

    const int n    = in_sizes[1];           // 16384 rows
    const int c    = in_sizes[0] / n;       // 4096 channels
    const int cvec = c / 4;                 // float4 vectors per row

    dim3 grid(n), block(THREADS);
    RandomShiftsAug_kernel<<<grid, block, 0, stream>>>(x, h1, w1, padp, out, cvec);
}